// NeighborhoodPool_5059471475303
// MI455X (gfx1250) — compile-verified
//
#include <hip/hip_runtime.h>
#include <math.h>

typedef float v2f __attribute__((ext_vector_type(2)));
typedef float v8f __attribute__((ext_vector_type(8)));

#define HIDC 64  // compile-time hidden width for the WMMA path (reference HID)

// ---------------- device helpers ----------------

static __device__ __forceinline__ float selu_f(float x) {
  const float scale = 1.0507009873554805f, alpha = 1.6732632423543772f;
  return scale * (x > 0.0f ? x : alpha * (__expf(x) - 1.0f));
}

// monotonic float <-> ordered-uint mapping (for atomicMax-based segment max)
static __device__ __forceinline__ unsigned f2ord(float f) {
  unsigned u = __float_as_uint(f);
  return (u & 0x80000000u) ? ~u : (u | 0x80000000u);
}
static __device__ __forceinline__ float ord2f(unsigned u) {
  return __uint_as_float((u & 0x80000000u) ? (u & 0x7FFFFFFFu) : ~u);
}

// decode augmented edge e in [0, 2E+N): fwd, reversed, self-loop
static __device__ __forceinline__ void aug_edge(int e, const int* __restrict__ e0,
                                                const int* __restrict__ e1, int E,
                                                int* s, int* d) {
  if (e < E)          { *s = e0[e];     *d = e1[e];     }
  else if (e < 2 * E) { *s = e1[e - E]; *d = e0[e - E]; }
  else                { *s = e - 2 * E; *d = e - 2 * E; }
}

// ---------------- generic fills ----------------

__global__ void fill_u32_kernel(unsigned* p, unsigned v, long long n) {
  long long i = (long long)blockIdx.x * blockDim.x + threadIdx.x;
  long long st = (long long)gridDim.x * blockDim.x;
  for (; i < n; i += st) p[i] = v;
}
__global__ void fill_f32_kernel(float* p, float v, long long n) {
  long long i = (long long)blockIdx.x * blockDim.x + threadIdx.x;
  long long st = (long long)gridDim.x * blockDim.x;
  for (; i < n; i += st) p[i] = v;
}
__global__ void copy_i32_kernel(int* dst, const int* src, int n) {
  int i = blockIdx.x * blockDim.x + threadIdx.x;
  if (i < n) dst[i] = src[i];
}

// ---------------- GCN scoring ----------------

__global__ void deg_kernel(float* deg, const int* e0, const int* e1, int E, int EAUG) {
  int e = blockIdx.x * blockDim.x + threadIdx.x;
  if (e >= EAUG) return;
  int s, d; aug_edge(e, e0, e1, E, &s, &d);
  atomicAdd(&deg[d], 1.0f);
}

__global__ void rsqrt_kernel(float* dinv, const float* deg, int N) {
  int i = blockIdx.x * blockDim.x + threadIdx.x;
  if (i < N) dinv[i] = rsqrtf(deg[i]);
}

__global__ void aggp_kernel(float* aggp, const float* pos, const float* dinv,
                            const int* e0, const int* e1, int E, int EAUG) {
  int e = blockIdx.x * blockDim.x + threadIdx.x;
  if (e >= EAUG) return;
  int s, d; aug_edge(e, e0, e1, E, &s, &d);
  float w = dinv[s] * dinv[d];
  atomicAdd(&aggp[d * 3 + 0], pos[s * 3 + 0] * w);
  atomicAdd(&aggp[d * 3 + 1], pos[s * 3 + 1] * w);
  atomicAdd(&aggp[d * 3 + 2], pos[s * 3 + 2] * w);
}

// h1 = selu(aggp[NPAD,3] @ W1[3,64] + b1) via V_WMMA_F32_16X16X4_F32.
// NPAD is a multiple of 128, so every wave owns one FULL 16-row tile:
// zero guards, 4 static WMMAs, straight-line stores.
__global__ __launch_bounds__(256) void gcn_dense1_wmma(
    const float* __restrict__ aggp, const float* __restrict__ W1,
    const float* __restrict__ b1, float* __restrict__ h1) {
  int lane = threadIdx.x & 31;
  int wave = threadIdx.x >> 5;
  int rowBase = (blockIdx.x * 8 + wave) * 16;
  int m  = lane & 15;
  bool hi = lane >= 16;                    // lanes 16-31 hold K=2,3 (A) / rows 2,3 (B)
  int r = rowBase + m;

  // A tile 16x4 (K padded 3->4 with zeros)
  v2f a;
  if (!hi) { a.x = aggp[r * 3 + 0]; a.y = aggp[r * 3 + 1]; }
  else     { a.x = aggp[r * 3 + 2]; a.y = 0.0f; }

#pragma unroll
  for (int t = 0; t < HIDC / 16; ++t) {
    int col = t * 16 + m;
    v2f b;
    if (!hi) { b.x = W1[0 * HIDC + col]; b.y = W1[1 * HIDC + col]; }
    else     { b.x = W1[2 * HIDC + col]; b.y = 0.0f; }
    v8f c = {};
    c = __builtin_amdgcn_wmma_f32_16x16x4_f32(
        /*neg_a=*/false, a, /*neg_b=*/false, b,
        /*c_mod=*/(short)0, c, /*reuse_a=*/false, /*reuse_b=*/false);
    float bias = b1[col];
    int rr0 = rowBase + (hi ? 8 : 0);      // D: lanes 16-31 hold M=8..15
#pragma unroll
    for (int v = 0; v < 8; ++v)
      h1[(rr0 + v) * HIDC + col] = selu_f(c[v] + bias);
  }
}

// scalar fallback for HID != 64
__global__ void gcn_dense1_scalar(const float* aggp, const float* W1,
                                  const float* b1, float* h1, int N, int HID) {
  long long idx = (long long)blockIdx.x * blockDim.x + threadIdx.x;
  if (idx >= (long long)N * HID) return;
  int i = (int)(idx / HID), h = (int)(idx - (long long)i * HID);
  float acc = b1[h];
  for (int k = 0; k < 3; ++k) acc += aggp[i * 3 + k] * W1[k * HID + h];
  h1[(long long)i * HID + h] = selu_f(acc);
}

__global__ void aggh_kernel(float* aggh, const float* h1, const float* dinv,
                            const int* e0, const int* e1, int E, int EAUG, int HID) {
  long long idx = (long long)blockIdx.x * blockDim.x + threadIdx.x;
  long long total = (long long)EAUG * HID;
  if (idx >= total) return;
  int e = (int)(idx / HID);
  int f = (int)(idx - (long long)e * HID);
  int s, d; aug_edge(e, e0, e1, E, &s, &d);
  float w = dinv[s] * dinv[d];
  atomicAdd(&aggh[d * HID + f], h1[s * HID + f] * w);
}

__global__ void score_kernel(const float* aggh, const float* W2, const float* b2,
                             float* score, float* out_s, int N, int HID) {
  int i = blockIdx.x * blockDim.x + threadIdx.x;
  if (i >= N) return;
  float acc = b2[0];
  for (int h = 0; h < HID; ++h) acc += aggh[i * HID + h] * W2[h];
  score[i] = acc;
  out_s[i] = acc;
}

// ---------------- CSR out-adjacency ----------------

__global__ void count_kernel(int* rowstart, const int* e0, const int* e1, int E, int EAUG) {
  int e = blockIdx.x * blockDim.x + threadIdx.x;
  if (e >= EAUG) return;
  int s, d; aug_edge(e, e0, e1, E, &s, &d);
  atomicAdd(&rowstart[s + 1], 1);
}
__global__ void scan_kernel(int* rowstart, int N) {
  if (blockIdx.x == 0 && threadIdx.x == 0)
    for (int i = 1; i <= N; ++i) rowstart[i] += rowstart[i - 1];
}
__global__ void adjfill_kernel(int* adj, int* cursor, const int* e0, const int* e1,
                               int E, int EAUG) {
  int e = blockIdx.x * blockDim.x + threadIdx.x;
  if (e >= EAUG) return;
  int s, d; aug_edge(e, e0, e1, E, &s, &d);
  int p = atomicAdd(&cursor[s], 1);
  adj[p] = d;
}

// ---------------- greedy clustering ----------------

// stable ascending rank by (score, index)
__global__ void rank_order_kernel(const float* __restrict__ score, int* order, int N) {
  int i = blockIdx.x * blockDim.x + threadIdx.x;
  if (i >= N) return;
  float si = score[i];
  int r = 0;
  for (int j = 0; j < N; ++j) {
    float sj = score[j];
    if (sj < si || (sj == si && j < i)) r++;
  }
  order[r] = i;
}

__global__ __launch_bounds__(256) void greedy_kernel(
    const int* __restrict__ order, const int* __restrict__ rowstart,
    const int* __restrict__ adj, int* assigned, int* cluster, int* seed,
    int* nout, int N) {
  __shared__ int s_node, s_ptr, s_n;
  int tid = threadIdx.x;
  if (tid == 0) { s_ptr = 0; s_n = 0; }
  __syncthreads();
  for (;;) {
    if (tid == 0) {
      int p = s_ptr;
      while (p < N && ((volatile int*)assigned)[order[p]]) p++;
      s_ptr = p;
      s_node = (p < N) ? order[p] : -1;
    }
    __syncthreads();
    int node = s_node;
    if (node < 0) break;
    int n = s_n;
    int beg = rowstart[node], end = rowstart[node + 1];
    for (int k = beg + tid; k < end; k += blockDim.x) {
      int v = adj[k];
      if (((volatile int*)assigned)[v] == 0) {   // includes self-loop -> assigns `node`
        assigned[v] = 1;
        cluster[v] = n;
      }
    }
    if (tid == 0) seed[n] = node;
    __threadfence();
    __syncthreads();
    if (tid == 0) { s_n = n + 1; s_ptr = s_ptr + 1; }
    __syncthreads();
  }
  if (tid == 0) *nout = s_n;
}

// ---------------- pooling ----------------

__global__ void counts_kernel(float* counts, const int* cluster, int N) {
  int i = blockIdx.x * blockDim.x + threadIdx.x;
  if (i < N) atomicAdd(&counts[cluster[i]], 1.0f);
}

__global__ void xpool_atomic_kernel(unsigned* pool, const float* __restrict__ x,
                                    const int* __restrict__ cluster, int N, int F) {
  long long idx = (long long)blockIdx.x * blockDim.x + threadIdx.x;
  long long total = (long long)N * F;
  if (idx >= total) return;
  int i = (int)(idx / F);
  int f = (int)(idx - (long long)i * F);
  atomicMax(&pool[(long long)cluster[i] * F + f], f2ord(x[idx]));
}

__global__ void xpool_final_kernel(float* out, const float* counts, int N, int F) {
  long long idx = (long long)blockIdx.x * blockDim.x + threadIdx.x;
  long long total = (long long)N * F;
  if (idx >= total) return;
  int c = (int)(idx / F);
  unsigned u = ((unsigned*)out)[idx];
  out[idx] = (counts[c] > 0.0f) ? ord2f(u) : 0.0f;
}

__global__ void pos_pool_kernel(float* out, const float* pos, const int* seed,
                                const int* nout, int N) {
  int idx = blockIdx.x * blockDim.x + threadIdx.x;
  if (idx >= N * 3) return;
  int c = idx / 3, d = idx - c * 3;
  out[idx] = (c < *nout) ? pos[seed[c] * 3 + d] : 0.0f;
}

// ---------------- pooled edge keys ----------------

__global__ void keys_init_kernel(int* keys, const int* cluster, const int* e0,
                                 const int* e1, int E, int M, int N, int SENT) {
  int i = blockIdx.x * blockDim.x + threadIdx.x;
  if (i >= M) return;
  int k = SENT;
  if (i < E) {
    int c0 = cluster[e0[i]], c1 = cluster[e1[i]];
    if (c0 != c1) k = c0 * N + c1;
  }
  keys[i] = k;
}

__global__ void bitonic_step_kernel(int* d, int M, int j, int k) {
  int i = blockIdx.x * blockDim.x + threadIdx.x;
  if (i >= M) return;
  int l = i ^ j;
  if (l > i) {
    int a = d[i], b = d[l];
    bool asc = ((i & k) == 0);
    bool swap = asc ? (a > b) : (a < b);
    if (swap) { d[i] = b; d[l] = a; }
  }
}

__global__ void uniq_kernel(const int* a, int* b, int M, int SENT) {
  int i = blockIdx.x * blockDim.x + threadIdx.x;
  if (i >= M) return;
  int v = a[i];
  b[i] = (v < SENT && (i == 0 || a[i - 1] != v)) ? v : SENT;
}

__global__ void emit_edges_kernel(const int* keys, int* out, int E) {
  int i = blockIdx.x * blockDim.x + threadIdx.x;
  if (i < E) out[i] = keys[i];
}

// ---------------- host launcher ----------------

extern "C" void kernel_launch(void* const* d_in, const int* in_sizes, int n_in,
                              void* d_out, int out_size, void* d_ws, size_t ws_size,
                              hipStream_t stream) {
  const float* x   = (const float*)d_in[0];
  const int*   ei  = (const int*)d_in[1];   // [2, E]
  const float* pos = (const float*)d_in[2]; // [N, 3]
  const float* W1  = (const float*)d_in[3]; // [3, HID]
  const float* b1  = (const float*)d_in[4]; // [HID]
  const float* W2  = (const float*)d_in[5]; // [HID, 1]
  const float* b2  = (const float*)d_in[6]; // [1]

  const int N    = in_sizes[2] / 3;
  const int F    = in_sizes[0] / N;
  const int E    = in_sizes[1] / 2;
  const int HID  = in_sizes[4];
  const int EAUG = 2 * E + N;
  const int SENT = N * N;
  const int NPAD = (N + 127) & ~127;        // full 16-row tiles for every wave
  int M = 1; while (M < E) M <<= 1;

  const int* e0 = ei;
  const int* e1 = ei + E;

  // output layout: x_pool [N*F] | edge_keys [E] (int32) | pos_pool [N*3] | score [N]
  float* out_x = (float*)d_out;
  int*   out_e = (int*)(out_x + (size_t)N * F);
  float* out_p = (float*)(out_e + E);
  float* out_s = out_p + (size_t)N * 3;

  // workspace carve-out
  char* wsb = (char*)d_ws; size_t off = 0;
  auto alloc = [&](size_t bytes) -> void* {
    void* p = wsb + off;
    off = (off + bytes + 255) & ~(size_t)255;
    return p;
  };
  float* deg      = (float*)alloc((size_t)N * 4);
  float* dinv     = (float*)alloc((size_t)N * 4);
  float* aggp     = (float*)alloc((size_t)NPAD * 3 * 4);
  float* h1       = (float*)alloc((size_t)NPAD * HID * 4);
  float* aggh     = (float*)alloc((size_t)N * HID * 4);
  float* score    = (float*)alloc((size_t)N * 4);
  int*   rowstart = (int*)alloc((size_t)(N + 1) * 4);
  int*   cursor   = (int*)alloc((size_t)N * 4);
  int*   adj      = (int*)alloc((size_t)EAUG * 4);
  int*   order    = (int*)alloc((size_t)N * 4);
  int*   assigned = (int*)alloc((size_t)N * 4);
  int*   cluster  = (int*)alloc((size_t)N * 4);
  int*   seed     = (int*)alloc((size_t)N * 4);
  int*   nout     = (int*)alloc(4);
  float* counts   = (float*)alloc((size_t)N * 4);
  int*   keysA    = (int*)alloc((size_t)M * 4);
  int*   keysB    = (int*)alloc((size_t)M * 4);

  const int T = 256;
  auto nb = [&](long long n) { return (int)((n + T - 1) / T); };
  const long long NF = (long long)N * F;

  // --- GCN layer 1 ---
  fill_f32_kernel<<<1024, T, 0, stream>>>(deg, 0.0f, N);
  deg_kernel<<<nb(EAUG), T, 0, stream>>>(deg, e0, e1, E, EAUG);
  rsqrt_kernel<<<nb(N), T, 0, stream>>>(dinv, deg, N);
  fill_f32_kernel<<<1024, T, 0, stream>>>(aggp, 0.0f, (long long)NPAD * 3); // incl. pad rows
  aggp_kernel<<<nb(EAUG), T, 0, stream>>>(aggp, pos, dinv, e0, e1, E, EAUG);
  if (HID == HIDC) {
    gcn_dense1_wmma<<<NPAD / 128, 256, 0, stream>>>(aggp, W1, b1, h1);
  } else {
    gcn_dense1_scalar<<<nb((long long)N * HID), T, 0, stream>>>(aggp, W1, b1, h1, N, HID);
  }

  // --- GCN layer 2 (score) ---
  fill_f32_kernel<<<1024, T, 0, stream>>>(aggh, 0.0f, (long long)N * HID);
  aggh_kernel<<<nb((long long)EAUG * HID), T, 0, stream>>>(aggh, h1, dinv, e0, e1, E, EAUG, HID);
  score_kernel<<<nb(N), T, 0, stream>>>(aggh, W2, b2, score, out_s, N, HID);

  // --- CSR out-adjacency on augmented edges ---
  fill_u32_kernel<<<256, T, 0, stream>>>((unsigned*)rowstart, 0u, N + 1);
  count_kernel<<<nb(EAUG), T, 0, stream>>>(rowstart, e0, e1, E, EAUG);
  scan_kernel<<<1, 1, 0, stream>>>(rowstart, N);
  copy_i32_kernel<<<nb(N), T, 0, stream>>>(cursor, rowstart, N);
  adjfill_kernel<<<nb(EAUG), T, 0, stream>>>(adj, cursor, e0, e1, E, EAUG);

  // --- greedy clustering ---
  rank_order_kernel<<<nb(N), T, 0, stream>>>(score, order, N);
  fill_u32_kernel<<<256, T, 0, stream>>>((unsigned*)assigned, 0u, N);
  fill_u32_kernel<<<256, T, 0, stream>>>((unsigned*)cluster, 0u, N);
  fill_u32_kernel<<<256, T, 0, stream>>>((unsigned*)seed, 0u, N);
  greedy_kernel<<<1, 256, 0, stream>>>(order, rowstart, adj, assigned, cluster, seed, nout, N);

  // --- per-cluster max pool of x (in d_out, ordered-uint atomicMax) ---
  fill_f32_kernel<<<1024, T, 0, stream>>>(counts, 0.0f, N);
  counts_kernel<<<nb(N), T, 0, stream>>>(counts, cluster, N);
  fill_u32_kernel<<<4096, T, 0, stream>>>((unsigned*)out_x, 0u, NF);
  xpool_atomic_kernel<<<nb(NF), T, 0, stream>>>((unsigned*)out_x, x, cluster, N, F);
  xpool_final_kernel<<<nb(NF), T, 0, stream>>>(out_x, counts, N, F);

  // --- pooled edge keys: sort, dedupe, re-sort ---
  keys_init_kernel<<<nb(M), T, 0, stream>>>(keysA, cluster, e0, e1, E, M, N, SENT);
  for (int k = 2; k <= M; k <<= 1)
    for (int j = k >> 1; j > 0; j >>= 1)
      bitonic_step_kernel<<<nb(M), T, 0, stream>>>(keysA, M, j, k);
  uniq_kernel<<<nb(M), T, 0, stream>>>(keysA, keysB, M, SENT);
  for (int k = 2; k <= M; k <<= 1)
    for (int j = k >> 1; j > 0; j >>= 1)
      bitonic_step_kernel<<<nb(M), T, 0, stream>>>(keysB, M, j, k);
  emit_edges_kernel<<<nb(E), T, 0, stream>>>(keysB, out_e, E);

  // --- pos_pool ---
  pos_pool_kernel<<<nb((long long)N * 3), T, 0, stream>>>(out_p, pos, seed, nout, N);
}